// MHCResidualWrapper_26877905338679
// MI455X (gfx1250) — compile-verified
//
#include <hip/hip_runtime.h>
#include <math.h>

typedef float v2f __attribute__((ext_vector_type(2)));
typedef float v8f __attribute__((ext_vector_type(8)));

#define SINKHORN_ITERS 10
#define TAU 0.05f

// ---------------------------------------------------------------------------
// Prep kernel: tiny 4x4 Sinkhorn + softmaxes -> Meff[i*4+j] in d_ws.
// Meff[i][j] = h_res[i][j] + h_post[i]*h_pre[j]; one thread, negligible cost.
// ---------------------------------------------------------------------------
__global__ void mhc_prep_kernel(const float* __restrict__ hres_logits,
                                const float* __restrict__ hpre_logits,
                                const float* __restrict__ hpost_logits,
                                float* __restrict__ Meff) {
    if (threadIdx.x != 0 || blockIdx.x != 0) return;

    float sc[16];
    float mx = -1e30f;
    for (int k = 0; k < 16; ++k) {
        sc[k] = hres_logits[k] / TAU;
        mx = fmaxf(mx, sc[k]);
    }
    for (int k = 0; k < 16; ++k) sc[k] -= mx;

    const float lm = -logf(4.0f);
    float u[4] = {0.f, 0.f, 0.f, 0.f};
    float v[4] = {0.f, 0.f, 0.f, 0.f};
    for (int it = 0; it < SINKHORN_ITERS; ++it) {
        for (int i = 0; i < 4; ++i) {            // u = lm - logsumexp_j(sc + v)
            float m = -1e30f;
            for (int j = 0; j < 4; ++j) m = fmaxf(m, sc[i * 4 + j] + v[j]);
            float s = 0.f;
            for (int j = 0; j < 4; ++j) s += expf(sc[i * 4 + j] + v[j] - m);
            u[i] = lm - (m + logf(s));
        }
        for (int j = 0; j < 4; ++j) {            // v = lm - logsumexp_i(sc + u)
            float m = -1e30f;
            for (int i = 0; i < 4; ++i) m = fmaxf(m, sc[i * 4 + j] + u[i]);
            float s = 0.f;
            for (int i = 0; i < 4; ++i) s += expf(sc[i * 4 + j] + u[i] - m);
            v[j] = lm - (m + logf(s));
        }
    }

    float pre[4], post[4];
    {
        float m = -1e30f;
        for (int k = 0; k < 4; ++k) m = fmaxf(m, hpre_logits[k]);
        float s = 0.f;
        for (int k = 0; k < 4; ++k) { pre[k] = expf(hpre_logits[k] - m); s += pre[k]; }
        for (int k = 0; k < 4; ++k) pre[k] /= s;
    }
    {
        float m = -1e30f;
        for (int k = 0; k < 4; ++k) m = fmaxf(m, hpost_logits[k]);
        float s = 0.f;
        for (int k = 0; k < 4; ++k) { post[k] = expf(hpost_logits[k] - m); s += post[k]; }
        for (int k = 0; k < 4; ++k) post[k] /= s;
    }

    for (int i = 0; i < 4; ++i)
        for (int j = 0; j < 4; ++j)
            Meff[i * 4 + j] = expf(sc[i * 4 + j] + u[i] + v[j]) * 4.0f + post[i] * pre[j];
}

// ---------------------------------------------------------------------------
// Main kernel: out(4xN) = Meff(4x4) * X(4xN) per stream-group b (= blockIdx.y).
// One V_WMMA_F32_16X16X4_F32 per 16-column tile.
//   A (16x4):  lane l, vgpr v  <- Meff[(l%16)&3][v + 2*(l/16)]   (tile-invariant)
//   B (4x16):  lane l, vgpr v  <- X[v + 2*(l/16)][n0 + l%16]
//   D:         vgpr r, lanes 0..15 = row r, col = lane  -> store rows 0..3
// Hot loop uses running pointers only (no per-iter index math) and
// unconditional speculative prefetch of the next tile.
// ---------------------------------------------------------------------------
__global__ void mhc_mix_wmma_kernel(const float* __restrict__ x,
                                    const float* __restrict__ M,
                                    float* __restrict__ out,
                                    long long N,            // T*D (multiple of 16)
                                    long long tilesPerB) {  // N/16
    const int lane = threadIdx.x & 31;
    const int half = lane >> 4;      // 0: K={0,1}, 1: K={2,3}
    const int nn   = lane & 15;      // column within tile
    const long long b = blockIdx.y;  // stream group 0..7

    const long long wavesPerB = ((long long)gridDim.x * blockDim.x) >> 5;
    const long long wave = (((long long)blockIdx.x * blockDim.x) + threadIdx.x) >> 5;

    // A operand: replicated mixing-matrix rows (C rows 4..15 are discarded)
    v2f a;
    a.x = M[(nn & 3) * 4 + 2 * half + 0];
    a.y = M[(nn & 3) * 4 + 2 * half + 1];

    const float* xg = x + (size_t)(b * 4) * (size_t)N;   // row j at xg + j*N
    float* og = out + (size_t)(b * 4) * (size_t)N;

    const long long strideElems = wavesPerB * 16;        // tile stride per wave

    long long iters = 0;
    if (wave < tilesPerB)
        iters = (tilesPerB - wave + wavesPerB - 1) / wavesPerB;

    // Running pointers (6 u64 adds per iteration, no index recomputation)
    const long long base = wave * 16 + nn;
    const float* pA = xg + (size_t)(2 * half) * (size_t)N + (size_t)base; // K = 2*half
    const float* pB = pA + (size_t)N;                                     // K = 2*half+1
    float* o0 = og + (size_t)base;
    float* o1 = o0 + (size_t)N;
    float* o2 = o0 + (size_t)N * 2;
    float* o3 = o0 + (size_t)N * 3;

    for (long long k = 0; k < iters; ++k) {
        v2f bv;
        bv.x = pA[0];
        bv.y = pB[0];

        // Speculative prefetch of next tile (safe past the end: dropped on
        // translation failure per ISA 10.5).
        __builtin_prefetch(pA + (size_t)strideElems, 0, 3);
        __builtin_prefetch(pB + (size_t)strideElems, 0, 3);

        v8f c = {};
        c = __builtin_amdgcn_wmma_f32_16x16x4_f32(
                /*neg_a=*/false, a, /*neg_b=*/false, bv,
                /*c_mod=*/(short)0, c, /*reuse_a=*/false, /*reuse_b=*/false);

        if (half == 0) {                 // lanes 0..15 carry rows 0..3 (real)
            *o0 = c[0];
            *o1 = c[1];
            *o2 = c[2];
            *o3 = c[3];
        }

        pA += (size_t)strideElems;
        pB += (size_t)strideElems;
        o0 += (size_t)strideElems;
        o1 += (size_t)strideElems;
        o2 += (size_t)strideElems;
        o3 += (size_t)strideElems;
    }

    // Generic tail (N % 16 != 0) — empty for the reference shapes.
    const long long tail0 = tilesPerB * 16;
    const long long tailN = N - tail0;
    if (tailN > 0) {
        const long long nThreads = (long long)gridDim.x * blockDim.x;
        const long long gtid = (long long)blockIdx.x * blockDim.x + threadIdx.x;
        float m00 = M[0], m01 = M[1], m02 = M[2], m03 = M[3];
        float m10 = M[4], m11 = M[5], m12 = M[6], m13 = M[7];
        float m20 = M[8], m21 = M[9], m22 = M[10], m23 = M[11];
        float m30 = M[12], m31 = M[13], m32 = M[14], m33 = M[15];
        for (long long e = gtid; e < tailN; e += nThreads) {
            const long long n2 = tail0 + e;
            float x0 = xg[n2], x1 = xg[(size_t)N + n2],
                  x2 = xg[(size_t)N * 2 + n2], x3 = xg[(size_t)N * 3 + n2];
            og[n2]                 = m00 * x0 + m01 * x1 + m02 * x2 + m03 * x3;
            og[(size_t)N + n2]     = m10 * x0 + m11 * x1 + m12 * x2 + m13 * x3;
            og[(size_t)N * 2 + n2] = m20 * x0 + m21 * x1 + m22 * x2 + m23 * x3;
            og[(size_t)N * 3 + n2] = m30 * x0 + m31 * x1 + m32 * x2 + m33 * x3;
        }
    }
}

extern "C" void kernel_launch(void* const* d_in, const int* in_sizes, int n_in,
                              void* d_out, int out_size, void* d_ws, size_t ws_size,
                              hipStream_t stream) {
    const float* streams      = (const float*)d_in[0];  // [32, T, D] fp32
    const float* hres_logits  = (const float*)d_in[1];  // [4,4]
    const float* hpre_logits  = (const float*)d_in[2];  // [4]
    const float* hpost_logits = (const float*)d_in[3];  // [4]
    float* out = (float*)d_out;
    float* Meff = (float*)d_ws;                         // 16 floats of scratch

    const long long total = (long long)in_sizes[0];     // 32 * T * D
    const long long N = total / 32;                     // T * D per stream row
    const long long tilesPerB = N / 16;

    mhc_prep_kernel<<<1, 32, 0, stream>>>(hres_logits, hpre_logits, hpost_logits, Meff);

    dim3 grid(512, 8, 1);   // y = stream group b; 4096 waves/group, 32 tiles/wave
    dim3 block(256, 1, 1);  // 8 waves (wave32)
    mhc_mix_wmma_kernel<<<grid, block, 0, stream>>>(streams, Meff, out, N, tilesPerB);
}